// ScaleDotProductAttention_43679817400327
// MI455X (gfx1250) — compile-verified
//
#include <hip/hip_runtime.h>
#include <hip/hip_bf16.h>
#include <math.h>

typedef __attribute__((ext_vector_type(2))) float v2f;
typedef __attribute__((ext_vector_type(8))) float v8f;
typedef __attribute__((ext_vector_type(4))) unsigned int u32x4;
typedef __attribute__((ext_vector_type(4))) int i32x4;
typedef __attribute__((ext_vector_type(8))) int i32x8;

#define ATT_B 4
#define ATT_H 16
#define ATT_S 2048
#define ATT_D 64
#define KTILES (ATT_S / 16)   // 128 key tiles of 16
#define DCHUNKS (ATT_D / 4)   // 16 chunks of K=4 for the f32 WMMA
#define WAVES_PER_BLOCK 4
#define ROW_PITCH 68          // 64 dwords + 4 pad dwords (TDM-inserted) -> no bank conflicts

__device__ __forceinline__ v8f wmma_f32(v2f a, v2f b, v8f c) {
    return __builtin_amdgcn_wmma_f32_16x16x4_f32(
        /*neg_a=*/false, a, /*neg_b=*/false, b,
        /*c_mod=*/(short)0, c, /*reuse_a=*/false, /*reuse_b=*/false);
}

// TDM: DMA one 16x64 fp32 tile (row stride 64 elems in memory) into LDS at
// lds_off, inserting 4 pad dwords after every 64 dwords (row pitch 68).
__device__ __forceinline__ void tdm_load_tile_16x64(const float* gptr, unsigned lds_off) {
    unsigned long long ga = (unsigned long long)(uintptr_t)gptr;
    u32x4 g0;
    g0[0] = 1u;                                   // count=1, user mode, no gather
    g0[1] = lds_off;                              // LDS byte address
    g0[2] = (unsigned)(ga & 0xffffffffu);         // global_addr[31:0]
    g0[3] = (unsigned)((ga >> 32) & 0x01ffffffu)  // global_addr[56:32]
          | (2u << 30);                           // type=2 ("image")
    i32x8 g1;
    g1[0] = (2 << 16)      // data_size = 4 bytes
          | (1 << 20)      // pad_enable
          | (5 << 22)      // pad_interval: 2^(5+1) = 64 dwords
          | (3 << 25);     // pad_amount: 3+1 = 4 dwords
    g1[1] = (ATT_D & 0xffff) << 16;               // tensor_dim0 = 64  (bits 79:48)
    g1[2] = (ATT_S & 0xffff) << 16;               // tensor_dim1 = 2048 (bits 111:80)
    g1[3] = (ATT_D & 0xffff) << 16;               // tile_dim0 = 64 (bits 127:112)
    g1[4] = 16;                                   // tile_dim1 = 16, tile_dim2 = 0
    g1[5] = ATT_D;                                // tensor_dim0_stride = 64 (low 32)
    g1[6] = 0;
    g1[7] = 0;
    i32x4 z4 = {0, 0, 0, 0};
#if __clang_major__ >= 23
    i32x8 z8 = {0, 0, 0, 0, 0, 0, 0, 0};
    __builtin_amdgcn_tensor_load_to_lds(g0, g1, z4, z4, z8, 0);
#else
    __builtin_amdgcn_tensor_load_to_lds(g0, g1, z4, z4, 0);
#endif
}

__global__ __launch_bounds__(WAVES_PER_BLOCK * 32)
void attn_fwd_wmma_f32(const float* __restrict__ qg,
                       const float* __restrict__ kg,
                       const float* __restrict__ vg,
                       const float* __restrict__ maskg,
                       float* __restrict__ ctxg,
                       float* __restrict__ attng) {
    const int wave = threadIdx.x >> 5;
    const int lane = threadIdx.x & 31;
    const int half = lane >> 4;       // 0: rows r, 1: rows r+8 in C layout
    const int ln   = lane & 15;       // fragment M / N index

    const int tile = blockIdx.x * WAVES_PER_BLOCK + wave;  // 0..8191
    const int qt = tile & (KTILES - 1);   // q-tile within (b,h); all 4 waves share bh
    const int bh = tile >> 7;             // b*H + h
    const int b  = bh >> 4;

    const float* Q = qg + (size_t)bh * ATT_S * ATT_D;
    const float* K = kg + (size_t)bh * ATT_S * ATT_D;
    const float* V = vg + (size_t)bh * ATT_S * ATT_D;
    const float* M = maskg + (size_t)b * ATT_S;
    float* C = ctxg  + (size_t)bh * ATT_S * ATT_D + (size_t)qt * 16 * ATT_D;
    float* A = attng + (size_t)bh * ATT_S * ATT_S + (size_t)qt * 16 * ATT_S;

    // Double-buffered TDM stages (shared by all 4 waves) + per-wave P relayout slab.
    __shared__ float ldsK[2][16 * ROW_PITCH];
    __shared__ float ldsV[2][16 * ROW_PITCH];
    __shared__ float ldsP[WAVES_PER_BLOCK][16][17];
    float (*P)[17] = ldsP[wave];

    const float scale = 0.125f;  // 1/sqrt(64)

    // ---- Preload Q A-fragments: chunk i covers d = 4i..4i+3.
    v2f qa[DCHUNKS];
#pragma unroll
    for (int i = 0; i < DCHUNKS; ++i) {
        const float* p = Q + ((size_t)(qt * 16 + ln)) * ATT_D + 4 * i + 2 * half;
        qa[i] = *(const v2f*)p;
    }

    // ================= Pass 1: online softmax stats (K staged via TDM) ========
    float m[8], s[8];
#pragma unroll
    for (int r = 0; r < 8; ++r) { m[r] = -INFINITY; s[r] = 0.0f; }

    if (wave == 0)
        tdm_load_tile_16x64(K, (unsigned)(uintptr_t)&ldsK[0][0]);

    for (int kt = 0; kt < KTILES; ++kt) {
        if (wave == 0) __builtin_amdgcn_s_wait_tensorcnt(0);
        __syncthreads();  // current buffer ready; alternate buffer fully consumed
        if (wave == 0 && kt + 1 < KTILES)
            tdm_load_tile_16x64(K + (size_t)(kt + 1) * 16 * ATT_D,
                                (unsigned)(uintptr_t)&ldsK[(kt + 1) & 1][0]);

        const float* kbuf = ldsK[kt & 1];
        v8f acc = {};
#pragma unroll
        for (int i = 0; i < DCHUNKS; ++i) {
            // B = K^T fragment: lane ln = key col, v0/v1 = d 2*half+{0,1}
            v2f kb = *(const v2f*)&kbuf[ln * ROW_PITCH + 4 * i + 2 * half];
            acc = wmma_f32(qa[i], kb, acc);
        }
        const float mk = M[kt * 16 + ln];
#pragma unroll
        for (int r = 0; r < 8; ++r) {
            float val = acc[r] * scale + mk;
            float tm = val;
#pragma unroll
            for (int off = 1; off < 16; off <<= 1)
                tm = fmaxf(tm, __shfl_xor(tm, off, 32));
            float mn = fmaxf(m[r], tm);
            float p  = __expf(val - mn);
            float ts = p;
#pragma unroll
            for (int off = 1; off < 16; off <<= 1)
                ts += __shfl_xor(ts, off, 32);
            s[r] = s[r] * __expf(m[r] - mn) + ts;
            m[r] = mn;
        }
    }

    float inv[8];
#pragma unroll
    for (int r = 0; r < 8; ++r) inv[r] = 1.0f / s[r];

    // ====== Pass 2: normalized attention out + P*V (K and V staged via TDM) ===
    v8f cacc[4];
#pragma unroll
    for (int n = 0; n < 4; ++n) cacc[n] = (v8f){};

    if (wave == 0) {
        tdm_load_tile_16x64(K, (unsigned)(uintptr_t)&ldsK[0][0]);
        tdm_load_tile_16x64(V, (unsigned)(uintptr_t)&ldsV[0][0]);
    }

    for (int kt = 0; kt < KTILES; ++kt) {
        if (wave == 0) __builtin_amdgcn_s_wait_tensorcnt(0);
        __syncthreads();
        if (wave == 0 && kt + 1 < KTILES) {
            tdm_load_tile_16x64(K + (size_t)(kt + 1) * 16 * ATT_D,
                                (unsigned)(uintptr_t)&ldsK[(kt + 1) & 1][0]);
            tdm_load_tile_16x64(V + (size_t)(kt + 1) * 16 * ATT_D,
                                (unsigned)(uintptr_t)&ldsV[(kt + 1) & 1][0]);
        }

        const float* kbuf = ldsK[kt & 1];
        const float* vbuf = ldsV[kt & 1];

        v8f acc = {};
#pragma unroll
        for (int i = 0; i < DCHUNKS; ++i) {
            v2f kb = *(const v2f*)&kbuf[ln * ROW_PITCH + 4 * i + 2 * half];
            acc = wmma_f32(qa[i], kb, acc);
        }
        const float mk = M[kt * 16 + ln];
#pragma unroll
        for (int r = 0; r < 8; ++r) {
            const int row = r + 8 * half;
            float pv = __expf(acc[r] * scale + mk - m[r]) * inv[r];
            A[(size_t)row * ATT_S + kt * 16 + ln] = pv;   // stream attention out
            P[row][ln] = pv;                               // stage for C->A relayout
        }
        asm volatile("s_wait_dscnt 0x0" ::: "memory");     // intra-wave LDS RAW

        // context += P(16x16) * V(16x16) per 16-wide d tile
#pragma unroll
        for (int j = 0; j < 4; ++j) {
            v2f pa;
            pa.x = P[ln][4 * j + 2 * half + 0];
            pa.y = P[ln][4 * j + 2 * half + 1];
#pragma unroll
            for (int n = 0; n < 4; ++n) {
                const int krow = 4 * j + 2 * half;
                v2f vb;
                vb.x = vbuf[(krow + 0) * ROW_PITCH + n * 16 + ln];
                vb.y = vbuf[(krow + 1) * ROW_PITCH + n * 16 + ln];
                cacc[n] = wmma_f32(pa, vb, cacc[n]);
            }
        }
    }

    // ---- Store context tile (16 x 64).
#pragma unroll
    for (int n = 0; n < 4; ++n)
#pragma unroll
        for (int r = 0; r < 8; ++r) {
            const int row = r + 8 * half;
            C[(size_t)row * ATT_D + n * 16 + ln] = cacc[n][r];
        }
}

extern "C" void kernel_launch(void* const* d_in, const int* in_sizes, int n_in,
                              void* d_out, int out_size, void* d_ws, size_t ws_size,
                              hipStream_t stream) {
    const float* q    = (const float*)d_in[0];
    const float* k    = (const float*)d_in[1];
    const float* v    = (const float*)d_in[2];
    const float* mask = (const float*)d_in[3];

    float* ctx  = (float*)d_out;                                   // [B,H,S,D]
    float* attn = ctx + (size_t)ATT_B * ATT_H * ATT_S * ATT_D;     // [B,H,S,S]

    const int total_tiles = ATT_B * ATT_H * (ATT_S / 16);          // 8192
    dim3 grid(total_tiles / WAVES_PER_BLOCK);
    dim3 block(WAVES_PER_BLOCK * 32);
    attn_fwd_wmma_f32<<<grid, block, 0, stream>>>(q, k, v, mask, ctx, attn);
}